// GraphSAGEEncoder_23210003268198
// MI455X (gfx1250) — compile-verified
//
#include <hip/hip_runtime.h>
#include <hip/hip_bf16.h>

// GraphSAGE encoder, MI455X (gfx1250, wave32).
// Memory-bound: ~1.3 GB of edge gather/scatter+activation traffic vs ~10 GFLOP
// of GEMM. Strategy: coalesced wave-per-edge scatter-add, fp32 WMMA
// (v_wmma_f32_16x16x4_f32) for the per-layer dual GEMM, fused bias/ReLU,
// separate wave-per-row LayerNorm with wave32 shuffle reductions.

#define NN 50000      // nodes
#define EE 800000     // edges
#define DD 128        // feature dim

typedef __attribute__((ext_vector_type(2))) float v2f;
typedef __attribute__((ext_vector_type(8))) float v8f;

// ---------------------------------------------------------------- zero agg
__global__ __launch_bounds__(256) void sage_zero_kernel(float4* __restrict__ p, int n4) {
    int i = blockIdx.x * 256 + threadIdx.x;
    if (i < n4) p[i] = make_float4(0.f, 0.f, 0.f, 0.f);
}

// ------------------------------------------------- edge gather + scatter-add
// One wave32 per edge: lane handles 4 contiguous floats -> the wave moves one
// full 512B feature row per edge with a single B128 load + 4 atomic f32 adds.
// src[e]/dst[e] are wave-uniform -> scalar loads.
__global__ __launch_bounds__(256) void sage_scatter_kernel(
    const float* __restrict__ X, const int* __restrict__ src,
    const int* __restrict__ dst, float* __restrict__ AG) {
    unsigned gid = blockIdx.x * 256u + threadIdx.x;
    unsigned e = gid >> 5;                 // edge id (uniform per wave)
    unsigned q = (gid & 31u) << 2;         // feature offset, 4 floats per lane
    int s = src[e];
    int d = dst[e];
    const float4 v = *(const float4*)(X + (size_t)s * DD + q);
    float* p = AG + (size_t)d * DD + q;
    atomicAdd(p + 0, v.x);
    atomicAdd(p + 1, v.y);
    atomicAdd(p + 2, v.z);
    atomicAdd(p + 3, v.w);
}

// ---------------------------------------------- fused dual-GEMM + bias (+relu)
// H = AG @ Wl^T + bl + X @ Wr^T ; W stored row-major [d_out, d_in].
// Block = 256 threads = 8 waves; block b owns 16 output rows, wave w owns
// output columns [16w,16w+16). Each wave runs K=128 in steps of 4 with two
// v_wmma_f32_16x16x4_f32 per step (one per GEMM) -> 64 WMMAs per tile.
__global__ __launch_bounds__(256) void sage_gemm_kernel(
    const float* __restrict__ X,   // root-path input  [NN,DD]
    const float* __restrict__ AG,  // aggregated input [NN,DD]
    const float* __restrict__ Wl,  // [DD,DD]
    const float* __restrict__ bl,  // [DD]
    const float* __restrict__ Wr,  // [DD,DD]
    float* __restrict__ H,         // [NN,DD]
    int relu) {
    const int wave = threadIdx.x >> 5;   // 0..7 -> column tile
    const int lane = threadIdx.x & 31;
    const int half = lane >> 4;          // 0/1: which K-pair (A) / K-pair (B)
    const int l16  = lane & 15;
    const int row0 = blockIdx.x * 16;    // 50000/16 = 3125 blocks, exact
    const int col0 = wave * 16;

    // A operand rows (per ISA: lanes 0-15 M=lane K={k,k+1}; lanes 16-31 K={k+2,k+3})
    const float* xrow  = X  + (size_t)(row0 + l16) * DD;
    const float* grow  = AG + (size_t)(row0 + l16) * DD;
    // B operand: B[k][n] = W[n][k]; lane supplies column n = col0+l16
    const float* wrrow = Wr + (size_t)(col0 + l16) * DD;
    const float* wlrow = Wl + (size_t)(col0 + l16) * DD;

    v8f acc = {0.f, 0.f, 0.f, 0.f, 0.f, 0.f, 0.f, 0.f};

#pragma unroll
    for (int k = 0; k < DD; k += 4) {
        const int ka = k + 2 * half;
        v2f aX = *(const v2f*)(xrow + ka);
        v2f bR = *(const v2f*)(wrrow + ka);
        acc = __builtin_amdgcn_wmma_f32_16x16x4_f32(
            /*neg_a=*/false, aX, /*neg_b=*/false, bR,
            /*c_mod=*/(short)0, acc, /*reuse_a=*/false, /*reuse_b=*/false);
        v2f aG = *(const v2f*)(grow + ka);
        v2f bL = *(const v2f*)(wlrow + ka);
        acc = __builtin_amdgcn_wmma_f32_16x16x4_f32(
            false, aG, false, bL, (short)0, acc, false, false);
    }

    // Epilogue: C/D layout -> VGPR r holds (row = r + 8*half, col = l16).
    const float bias = bl[col0 + l16];
    float* hout = H + (size_t)row0 * DD + col0 + l16;
#pragma unroll
    for (int r = 0; r < 8; ++r) {
        float v = acc[r] + bias;
        if (relu) v = fmaxf(v, 0.f);
        hout[(size_t)(r + 8 * half) * DD] = v;
    }
}

// ------------------------------------------------------- LayerNorm + ReLU
// One wave32 per row (4 floats/lane), wave32 shuffle tree reduction.
__global__ __launch_bounds__(256) void sage_lnrelu_kernel(
    float* __restrict__ H, const float* __restrict__ g, const float* __restrict__ b) {
    const int row  = blockIdx.x * 8 + (threadIdx.x >> 5);  // 50000/8 = 6250 blocks
    const int lane = threadIdx.x & 31;
    float* p = H + (size_t)row * DD + lane * 4;
    float4 v = *(const float4*)p;
    float s  = v.x + v.y + v.z + v.w;
    float s2 = v.x * v.x + v.y * v.y + v.z * v.z + v.w * v.w;
#pragma unroll
    for (int m = 16; m >= 1; m >>= 1) {
        s  += __shfl_xor(s, m, 32);
        s2 += __shfl_xor(s2, m, 32);
    }
    const float mu  = s * (1.f / DD);
    const float var = s2 * (1.f / DD) - mu * mu;
    const float rs  = rsqrtf(var + 1e-5f);
    const float4 gg = *(const float4*)(g + lane * 4);
    const float4 bb = *(const float4*)(b + lane * 4);
    float4 o;
    o.x = fmaxf((v.x - mu) * rs * gg.x + bb.x, 0.f);
    o.y = fmaxf((v.y - mu) * rs * gg.y + bb.y, 0.f);
    o.z = fmaxf((v.z - mu) * rs * gg.z + bb.z, 0.f);
    o.w = fmaxf((v.w - mu) * rs * gg.w + bb.w, 0.f);
    *(float4*)p = o;
}

extern "C" void kernel_launch(void* const* d_in, const int* in_sizes, int n_in,
                              void* d_out, int out_size, void* d_ws, size_t ws_size,
                              hipStream_t stream) {
    // setup_inputs order:
    // 0:x 1:edge_index 2:Wl0 3:bl0 4:Wr0 5:Wl1 6:bl1 7:Wr1 8:Wl2 9:bl2 10:Wr2 11:ln_g 12:ln_b
    const float* x    = (const float*)d_in[0];
    const int*   ei   = (const int*)d_in[1];
    const int*   src  = ei;           // edge_index[0]
    const int*   dst  = ei + EE;      // edge_index[1]
    const float* Wl0  = (const float*)d_in[2];
    const float* bl0  = (const float*)d_in[3];
    const float* Wr0  = (const float*)d_in[4];
    const float* Wl1  = (const float*)d_in[5];
    const float* bl1  = (const float*)d_in[6];
    const float* Wr1  = (const float*)d_in[7];
    const float* Wl2  = (const float*)d_in[8];
    const float* bl2  = (const float*)d_in[9];
    const float* Wr2  = (const float*)d_in[10];
    const float* ln_g = (const float*)d_in[11];
    const float* ln_b = (const float*)d_in[12];
    float* out = (float*)d_out;

    float* ws  = (float*)d_ws;
    float* agg = ws;                           // [NN*DD]
    float* hA  = ws + (size_t)NN * DD;         // [NN*DD]
    float* hB  = hA + (size_t)NN * DD;         // [NN*DD]

    const dim3 blk(256);
    const int gZero = (NN * DD / 4 + 255) / 256;   // zero agg (float4)
    const int gScat = (EE * 32) / 256;             // 100000, exact
    const int gGemm = NN / 16;                     // 3125, exact
    const int gLn   = NN / 8;                      // 6250, exact

    // ---- layer 0: conv -> layernorm -> relu
    sage_zero_kernel<<<gZero, blk, 0, stream>>>((float4*)agg, NN * DD / 4);
    sage_scatter_kernel<<<gScat, blk, 0, stream>>>(x, src, dst, agg);
    sage_gemm_kernel<<<gGemm, blk, 0, stream>>>(x, agg, Wl0, bl0, Wr0, hA, /*relu=*/0);
    sage_lnrelu_kernel<<<gLn, blk, 0, stream>>>(hA, ln_g, ln_b);

    // ---- layer 1: conv -> relu
    sage_zero_kernel<<<gZero, blk, 0, stream>>>((float4*)agg, NN * DD / 4);
    sage_scatter_kernel<<<gScat, blk, 0, stream>>>(hA, src, dst, agg);
    sage_gemm_kernel<<<gGemm, blk, 0, stream>>>(hA, agg, Wl1, bl1, Wr1, hB, /*relu=*/1);

    // ---- layer 2: final conv
    sage_zero_kernel<<<gZero, blk, 0, stream>>>((float4*)agg, NN * DD / 4);
    sage_scatter_kernel<<<gScat, blk, 0, stream>>>(hB, src, dst, agg);
    sage_gemm_kernel<<<gGemm, blk, 0, stream>>>(hB, agg, Wl2, bl2, Wr2, out, /*relu=*/0);
}